// MovingNormalizationLayer_15109694948102
// MI455X (gfx1250) — compile-verified
//
#include <hip/hip_runtime.h>

#ifndef __has_builtin
#define __has_builtin(x) 0
#endif

#define ROW_W 512
#define LANES 32
#define ROWS_PER_BLOCK 8
#define THREADS (ROWS_PER_BLOCK * LANES)
#define WIN_LO 50   // elements before center (pad left)
#define WIN_HI 49   // elements after center  (pad right)

struct RowSlot {
    float x[ROW_W];        // raw row data (filled by async global->LDS DMA)
    float P[ROW_W + 4];    // prefix sums of x   (P[0] = 0)
    float Q[ROW_W + 4];    // prefix sums of x^2 (Q[0] = 0)
};

// ---- CDNA5 async global->LDS (ASYNCcnt-tracked), 16B per lane ----
typedef int v4i __attribute__((vector_size(16)));
typedef __attribute__((address_space(1))) v4i* gv4i_p;   // global int4*
typedef __attribute__((address_space(3))) v4i* lv4i_p;   // LDS    int4*

__device__ __forceinline__ void async_g2l_b128(const float* gsrc, float* ldst) {
#if __has_builtin(__builtin_amdgcn_global_load_async_to_lds_b128)
    __builtin_amdgcn_global_load_async_to_lds_b128(
        (gv4i_p)(unsigned long long)gsrc,
        (lv4i_p)(unsigned)(unsigned long long)ldst,
        0, 0);
#else
    unsigned           l32 = (unsigned)(unsigned long long)ldst;   // LDS byte offset
    unsigned long long g64 = (unsigned long long)gsrc;
    asm volatile("global_load_async_to_lds_b128 %0, %1, off"
                 :: "v"(l32), "v"(g64) : "memory");
#endif
}

__device__ __forceinline__ void wait_async_zero() {
#if __has_builtin(__builtin_amdgcn_s_wait_asynccnt)
    __builtin_amdgcn_s_wait_asynccnt(0);
#else
    asm volatile("s_wait_asynccnt 0" ::: "memory");
#endif
}

__global__ __launch_bounds__(THREADS)
void moving_norm_kernel(const float* __restrict__ x,
                        float* __restrict__ out,
                        int nrows) {
    __shared__ RowSlot slots[ROWS_PER_BLOCK];

    const int lane = threadIdx.x & (LANES - 1);
    const int wave = threadIdx.x >> 5;                    // wave32: 8 waves/block
    const int row  = blockIdx.x * ROWS_PER_BLOCK + wave;
    if (row >= nrows) return;                             // whole-wave exit

    RowSlot& s = slots[wave];
    const float* grow = x + (size_t)row * ROW_W;

    // ---------- Phase 0: DMA row (2 KB) into LDS, 4 x b128 per lane-set ----------
#pragma unroll
    for (int j = 0; j < 4; ++j) {
        const int e = j * 128 + lane * 4;                 // coalesced: 32 lanes x 16B
        async_g2l_b128(grow + e, &s.x[e]);
    }
    wait_async_zero();                                    // ASYNCcnt -> 0 (wave-local slot)

    // ---------- Phase 1: prefix sums of x and x^2 ----------
    const int base = lane * 16;                           // 512 / 32 lanes
    float px[16], qx[16];
    float sx = 0.0f, sq = 0.0f;
#pragma unroll
    for (int k = 0; k < 16; ++k) {
        const float v = s.x[base + k];
        sx += v;       px[k] = sx;
        sq += v * v;   qx[k] = sq;
    }

    // wave32 exclusive scan of per-lane totals (5 shuffle steps)
    float tx = sx, tq = sq;
#pragma unroll
    for (int off = 1; off < LANES; off <<= 1) {
        const float ux = __shfl_up(tx, off, LANES);
        const float uq = __shfl_up(tq, off, LANES);
        if (lane >= off) { tx += ux; tq += uq; }
    }
    const float ex = tx - sx;                             // exclusive prefix of totals
    const float eq = tq - sq;

    if (lane == 0) { s.P[0] = 0.0f; s.Q[0] = 0.0f; }
#pragma unroll
    for (int k = 0; k < 16; ++k) {
        s.P[base + k + 1] = ex + px[k];
        s.Q[base + k + 1] = eq + qx[k];
    }
    // producer == consumer wave for this slot; compiler inserts s_wait_dscnt for RAW.

    // ---------- Phase 2: windowed normalize via prefix differences ----------
    float* orow = out + (size_t)row * ROW_W;
#pragma unroll
    for (int j = 0; j < 4; ++j) {
        const int e0 = j * 128 + lane * 4;                // coalesced float4 stores
        float r[4];
#pragma unroll
        for (int k = 0; k < 4; ++k) {
            const int i  = e0 + k;
            int lo = i - WIN_LO; if (lo < 0) lo = 0;
            int hi = i + WIN_HI; if (hi > ROW_W - 1) hi = ROW_W - 1;
            const float cnt  = (float)(hi - lo + 1);      // edge-aware norm (box of ones)
            const float sum  = s.P[hi + 1] - s.P[lo];
            const float sum2 = s.Q[hi + 1] - s.Q[lo];
            const float mean = sum / cnt;
            float var = sum2 / cnt - mean * mean;
            var = fmaxf(var, 1e-10f);                     // clip(a_min=1e-10)
            float r0 = (s.x[i] - mean) / sqrtf(var);
            if (r0 != r0) r0 = 0.0f;                      // nan_to_num: nan -> 0
            r0 = fminf(fmaxf(r0, -3.4028234663852886e38f),
                        3.4028234663852886e38f);          // +/-inf -> +/-FLT_MAX
            r[k] = r0;
        }
        float4 v4; v4.x = r[0]; v4.y = r[1]; v4.z = r[2]; v4.w = r[3];
        ((float4*)orow)[j * 32 + lane] = v4;              // global_store_b128
    }
}

extern "C" void kernel_launch(void* const* d_in, const int* in_sizes, int n_in,
                              void* d_out, int out_size, void* d_ws, size_t ws_size,
                              hipStream_t stream) {
    (void)n_in; (void)out_size; (void)d_ws; (void)ws_size;
    const float* x = (const float*)d_in[0];
    float* out = (float*)d_out;

    const int total = in_sizes[0];                        // 2*256*256*512 = 67,108,864
    const int nrows = total / ROW_W;                      // 131,072 rows
    const int blocks = (nrows + ROWS_PER_BLOCK - 1) / ROWS_PER_BLOCK;

    moving_norm_kernel<<<blocks, THREADS, 0, stream>>>(x, out, nrows);
}